// PSDDNBboxLoss_43757126812326
// MI455X (gfx1250) — compile-verified
//
#include <hip/hip_runtime.h>
#include <hip/hip_bf16.h>
#include <float.h>

typedef float v2f __attribute__((ext_vector_type(2)));
typedef float v8f __attribute__((ext_vector_type(8)));

#define NSEG 11
#define BLOCK 256

// d_ws float layout:
//  [0] center_acc  [1] size_acc  [2] ymin bits (u32)  [3] ymax bits (u32)
//  [8..63)   band accumulators: cnt[11], s1w[11], s1h[11], s2w[11], s2h[11]
//  [64..119) band params: ubw[11], ubh[11], lbw[11], lbh[11], valid[11]
//  [128..128+N) weight[i]   (only if ws_size permits)

__global__ void k_init(float* __restrict__ ws) {
    int t = threadIdx.x;
    unsigned int* wsu = (unsigned int*)ws;
    if (t == 2)      wsu[2] = 0x7F7FFFFFu;   // +FLT_MAX bits (min identity, y >= 0)
    else if (t == 3) wsu[3] = 0x00000000u;   // +0.0f bits   (max identity, y >= 0)
    else if (t < 128) ws[t] = 0.0f;
}

// Pass 1: WMMA row-sum of 80 scores -> weight; center loss partial; y min/max.
__global__ void __launch_bounds__(BLOCK)
k_weight_center(const float4* __restrict__ pb, const float4* __restrict__ tb,
                const float* __restrict__ scores, const int* __restrict__ fg,
                float* __restrict__ ws, int storeW, int N) {
    __shared__ float s_w[BLOCK];
    __shared__ float s_red[BLOCK];
    __shared__ float s_min[BLOCK];
    __shared__ float s_max[BLOCK];

    const int t    = threadIdx.x;
    const int wave = t >> 5;
    const int lane = t & 31;
    const int half = lane >> 4;       // K-half per 16x4 F32 A layout
    const int m    = lane & 15;       // row within tile
    const long long r0 = (long long)blockIdx.x * BLOCK + (long long)wave * 32;

    if (r0 + 32 <= (long long)N) {
        // Two 16-row tiles per wave; A loads match ISA A-layout:
        // lanes 0-15 hold K = c..c+1, lanes 16-31 hold K = c+2..c+3.
        const float* rowA = scores + ((r0 + m) * 80 + 2 * half);
        const float* rowB = scores + ((r0 + 16 + m) * 80 + 2 * half);
        v8f c0 = {}; v8f c1 = {};
        v2f b; b.x = 1.0f; b.y = 1.0f;            // B = ones(4x16)
#pragma unroll
        for (int c = 0; c < 80; c += 4) {
            v2f a0 = *(const v2f*)(rowA + c);
            v2f a1 = *(const v2f*)(rowB + c);
            c0 = __builtin_amdgcn_wmma_f32_16x16x4_f32(false, a0, false, b, (short)0, c0, false, false);
            c1 = __builtin_amdgcn_wmma_f32_16x16x4_f32(false, a1, false, b, (short)0, c1, false, false);
        }
        // D column 0 lives in lanes 0 (M=0..7) and 16 (M=8..15), VGPRs 0..7.
        if (m == 0) {
#pragma unroll
            for (int j = 0; j < 8; ++j) {
                s_w[wave * 32 + half * 8 + j]      = c0[j];
                s_w[wave * 32 + 16 + half * 8 + j] = c1[j];
            }
        }
    } else {
        // tail fallback (not hit for N = 1,075,200): scalar row sum
        long long i = r0 + lane;
        float s = 0.0f;
        if (i < N) {
            const float4* r = (const float4*)(scores + i * 80);
#pragma unroll
            for (int c = 0; c < 20; ++c) { float4 v = r[c]; s += v.x + v.y + v.z + v.w; }
        }
        s_w[t] = s;
    }
    __syncthreads();

    long long i = (long long)blockIdx.x * BLOCK + t;
    float cterm = 0.0f, ymin = FLT_MAX, ymax = -FLT_MAX;
    if (i < N) {
        float4 p  = pb[i];
        float4 tt = tb[i];
        int f = (fg[i] != 0);
        float wgt = f ? s_w[t] : 0.0f;
        if (storeW) ws[128 + i] = wgt;
        float pcx = 0.5f * (p.x + p.z),  pcy = 0.5f * (p.y + p.w);
        float tcx = 0.5f * (tt.x + tt.z), tcy = 0.5f * (tt.y + tt.w);
        float dx = pcx - tcx, dy = pcy - tcy;
        cterm = (dx * dx + dy * dy) * wgt;
        if (f) { ymin = tcy; ymax = tcy; }
    }
    s_red[t] = cterm; s_min[t] = ymin; s_max[t] = ymax;
    __syncthreads();
    for (int s = BLOCK / 2; s > 0; s >>= 1) {
        if (t < s) {
            s_red[t] += s_red[t + s];
            s_min[t]  = fminf(s_min[t], s_min[t + s]);
            s_max[t]  = fmaxf(s_max[t], s_max[t + s]);
        }
        __syncthreads();
    }
    if (t == 0) {
        atomicAdd(&ws[0], s_red[0]);
        if (s_min[0] <  FLT_MAX) atomicMin((unsigned int*)&ws[2], __float_as_uint(s_min[0]));
        if (s_max[0] > -FLT_MAX) atomicMax((unsigned int*)&ws[3], __float_as_uint(s_max[0]));
    }
}

__device__ __forceinline__ int band_of(float y, float ymin, float ymax) {
    float range = ymax - ymin;
    float yn = (y - ymin) / (range + 1e-6f);
    int band = (int)(yn * 10.0f);              // trunc toward zero, like astype(int32)
    band = min(max(band, 0), 9);
    if (range < 1e-6f) band = 0;
    return band;
}

// Pass 2: per-band segment sums (fg only contributes; fgf==0 otherwise).
__global__ void __launch_bounds__(BLOCK)
k_band_acc(const float4* __restrict__ tb, const int* __restrict__ fg,
           float* __restrict__ ws, int N) {
    __shared__ float acc[5 * NSEG];
    int t = threadIdx.x;
    if (t < 5 * NSEG) acc[t] = 0.0f;
    __syncthreads();
    long long i = (long long)blockIdx.x * BLOCK + t;
    if (i < N && fg[i] != 0) {
        float ymin = __uint_as_float(((const unsigned int*)ws)[2]);
        float ymax = __uint_as_float(((const unsigned int*)ws)[3]);
        float4 tt = tb[i];
        float tcy = 0.5f * (tt.y + tt.w);
        float tw = tt.z - tt.x, th = tt.w - tt.y;
        int band = band_of(tcy, ymin, ymax);
        atomicAdd(&acc[band],            1.0f);
        atomicAdd(&acc[NSEG + band],     tw);
        atomicAdd(&acc[2 * NSEG + band], th);
        atomicAdd(&acc[3 * NSEG + band], tw * tw);
        atomicAdd(&acc[4 * NSEG + band], th * th);
    }
    __syncthreads();
    if (t < 5 * NSEG && acc[t] != 0.0f) atomicAdd(&ws[8 + t], acc[t]);
}

// Pass 3: tiny — band stats -> ub/lb/valid.
__global__ void k_band_fin(float* __restrict__ ws) {
    int t = threadIdx.x;
    if (t < NSEG) {
        float cnt = ws[8 + t];
        float s1w = ws[8 + NSEG + t],     s1h = ws[8 + 2 * NSEG + t];
        float s2w = ws[8 + 3 * NSEG + t], s2h = ws[8 + 4 * NSEG + t];
        float n = fmaxf(cnt, 1.0f);
        float mw = s1w / n, mh = s1h / n;
        float d = fmaxf(cnt - 1.0f, 1.0f);
        float vw = fmaxf(s2w - cnt * mw * mw, 0.0f) / d;
        float vh = fmaxf(s2h - cnt * mh * mh, 0.0f) / d;
        float sw = sqrtf(vw) + 1e-6f, sh = sqrtf(vh) + 1e-6f;
        ws[64 + t]            = mw + 3.0f * sw;          // ubw
        ws[64 + NSEG + t]     = mh + 3.0f * sh;          // ubh
        ws[64 + 2 * NSEG + t] = fmaxf(mw - 3.0f * sw, 0.0f); // lbw
        ws[64 + 3 * NSEG + t] = fmaxf(mh - 3.0f * sh, 0.0f); // lbh
        ws[64 + 4 * NSEG + t] = (cnt >= 2.0f) ? 1.0f : 0.0f; // valid
    }
}

// Pass 4: size loss (weight==0 for non-fg, so only fg elements contribute).
__global__ void __launch_bounds__(BLOCK)
k_size(const float4* __restrict__ pb, const float4* __restrict__ tb,
       const float* __restrict__ scores, const int* __restrict__ fg,
       float* __restrict__ ws, int storeW, int N) {
    __shared__ float s_par[5 * NSEG];
    __shared__ float s_red[BLOCK];
    int t = threadIdx.x;
    if (t < 5 * NSEG) s_par[t] = ws[64 + t];
    __syncthreads();
    long long i = (long long)blockIdx.x * BLOCK + t;
    float term = 0.0f;
    if (i < N && fg[i] != 0) {
        float wgt;
        if (storeW) {
            wgt = ws[128 + i];
        } else {
            float s = 0.0f;
            const float4* r = (const float4*)(scores + i * 80);
#pragma unroll
            for (int c = 0; c < 20; ++c) { float4 v = r[c]; s += v.x + v.y + v.z + v.w; }
            wgt = s;
        }
        float ymin = __uint_as_float(((const unsigned int*)ws)[2]);
        float ymax = __uint_as_float(((const unsigned int*)ws)[3]);
        float4 tt = tb[i];
        int band = band_of(0.5f * (tt.y + tt.w), ymin, ymax);
        float ubw = s_par[band],            ubh = s_par[NSEG + band];
        float lbw = s_par[2 * NSEG + band], lbh = s_par[3 * NSEG + band];
        float vld = s_par[4 * NSEG + band];
        float4 p = pb[i];
        float pw = p.z - p.x, ph = p.w - p.y;
        float lw = (pw > ubw) ? (pw - ubw) * (pw - ubw) : 0.0f;
        lw      += (pw < lbw) ? (lbw - pw) * (lbw - pw) : 0.0f;
        float lh = (ph > ubh) ? (ph - ubh) * (ph - ubh) : 0.0f;
        lh      += (ph < lbh) ? (lbh - ph) * (lbh - ph) : 0.0f;
        term = (lw + lh) * vld * wgt;
    }
    s_red[t] = term;
    __syncthreads();
    for (int s = BLOCK / 2; s > 0; s >>= 1) {
        if (t < s) s_red[t] += s_red[t + s];
        __syncthreads();
    }
    if (t == 0) atomicAdd(&ws[1], s_red[0]);
}

__global__ void k_final(const float* __restrict__ tss, const float* __restrict__ ws,
                        float* __restrict__ out) {
    if (threadIdx.x == 0) {
        float s = tss[0];
        out[0] = (ws[0] + ws[1]) / s;   // loss_bbox
        out[1] = 0.0f;                  // loss_dfl
    }
}

extern "C" void kernel_launch(void* const* d_in, const int* in_sizes, int n_in,
                              void* d_out, int out_size, void* d_ws, size_t ws_size,
                              hipStream_t stream) {
    // inputs: 0 pred_dist (unused), 1 pred_bboxes, 2 anchor_points (unused),
    //         3 target_bboxes, 4 target_scores, 5 target_scores_sum, 6 fg_mask
    const float4* pb     = (const float4*)d_in[1];
    const float4* tb     = (const float4*)d_in[3];
    const float*  scores = (const float*)d_in[4];
    const float*  tss    = (const float*)d_in[5];
    const int*    fg     = (const int*)d_in[6];
    const int N = in_sizes[6];                // B*A = 1,075,200
    float* ws = (float*)d_ws;

    const size_t needed = (size_t)(128 + N) * sizeof(float);
    const int storeW = (ws_size >= needed) ? 1 : 0;
    const int blocks = (N + BLOCK - 1) / BLOCK;

    k_init<<<1, 128, 0, stream>>>(ws);
    k_weight_center<<<blocks, BLOCK, 0, stream>>>(pb, tb, scores, fg, ws, storeW, N);
    k_band_acc<<<blocks, BLOCK, 0, stream>>>(tb, fg, ws, N);
    k_band_fin<<<1, 64, 0, stream>>>(ws);
    k_size<<<blocks, BLOCK, 0, stream>>>(pb, tb, scores, fg, ws, storeW, N);
    k_final<<<1, 32, 0, stream>>>(tss, ws, (float*)d_out);
}